// AttnProcessor_80135499809226
// MI455X (gfx1250) — compile-verified
//
#include <hip/hip_runtime.h>
#include <math.h>

typedef __attribute__((ext_vector_type(2))) float v2f;
typedef __attribute__((ext_vector_type(8))) float v8f;

#define Bq 4
#define Sq 1024
#define Cq 640
#define Hq 8
#define Dq 80
#define CDq 1280
#define GEN_COND 2
#define REF_COND 3

// ---------------------------------------------------------------------------
// QKV projection: Y = X @ W^T, X:[4096,640], W:[640,640].
// One wave per 32x32 output block (2x2 register tiles): 4 WMMAs per 4 b64
// loads -> 8 FLOP/byte arithmetic intensity.
// grid: (128, 5, 3)  block: 128 (4 waves; wave picks n-block)
// ---------------------------------------------------------------------------
__global__ __launch_bounds__(128) void qkv_proj_kernel(
    const float* __restrict__ X,
    const float* __restrict__ Wq, const float* __restrict__ Wk,
    const float* __restrict__ Wv,
    float* __restrict__ qb, float* __restrict__ kb, float* __restrict__ vb)
{
    const float* W = (blockIdx.z == 0) ? Wq : ((blockIdx.z == 1) ? Wk : Wv);
    float*       Y = (blockIdx.z == 0) ? qb : ((blockIdx.z == 1) ? kb : vb);

    const int wave = threadIdx.x >> 5;
    const int lane = threadIdx.x & 31;
    const int lh   = lane & 15;
    const int hf   = lane >> 4;          // 0: lanes 0-15, 1: lanes 16-31

    const int m0 = blockIdx.x * 32;
    const int n0 = (blockIdx.y * 4 + wave) * 32;

    v8f acc00 = {}, acc01 = {}, acc10 = {}, acc11 = {};

    const float* xp0 = X + (size_t)(m0 + lh) * Cq + 2 * hf;
    const float* xp1 = xp0 + (size_t)16 * Cq;
    const float* wp0 = W + (size_t)(n0 + lh) * Cq + 2 * hf;
    const float* wp1 = wp0 + (size_t)16 * Cq;

#pragma unroll 4
    for (int k0 = 0; k0 < Cq; k0 += 4) {
        v2f a0 = *(const v2f*)(xp0 + k0);
        v2f a1 = *(const v2f*)(xp1 + k0);
        v2f b0 = *(const v2f*)(wp0 + k0);
        v2f b1 = *(const v2f*)(wp1 + k0);
        acc00 = __builtin_amdgcn_wmma_f32_16x16x4_f32(
                    false, a0, false, b0, (short)0, acc00, false, false);
        acc01 = __builtin_amdgcn_wmma_f32_16x16x4_f32(
                    false, a0, false, b1, (short)0, acc01, false, false);
        acc10 = __builtin_amdgcn_wmma_f32_16x16x4_f32(
                    false, a1, false, b0, (short)0, acc10, false, false);
        acc11 = __builtin_amdgcn_wmma_f32_16x16x4_f32(
                    false, a1, false, b1, (short)0, acc11, false, false);
    }

#pragma unroll
    for (int r = 0; r < 8; ++r) {
        int row = r + 8 * hf;
        Y[(size_t)(m0 + row)      * Cq + n0 + lh]      = acc00[r];
        Y[(size_t)(m0 + row)      * Cq + n0 + 16 + lh] = acc01[r];
        Y[(size_t)(m0 + 16 + row) * Cq + n0 + lh]      = acc10[r];
        Y[(size_t)(m0 + 16 + row) * Cq + n0 + 16 + lh] = acc11[r];
    }
}

// ---------------------------------------------------------------------------
// Inverse norms: rows 0..1023 -> ref, 1024..2047 -> tgt.  grid 2048, block 256
// ---------------------------------------------------------------------------
__global__ __launch_bounds__(256) void inv_norm_kernel(
    const float* __restrict__ ref, const float* __restrict__ tgt,
    float* __restrict__ rinv, float* __restrict__ tinv)
{
    __shared__ float red[256];
    const int row = blockIdx.x;
    const float* src = (row < Sq) ? (ref + (size_t)row * CDq)
                                  : (tgt + (size_t)(row - Sq) * CDq);
    float s = 0.f;
    for (int c = threadIdx.x; c < CDq; c += 256) s += src[c] * src[c];
    red[threadIdx.x] = s;
    __syncthreads();
    for (int st = 128; st > 0; st >>= 1) {
        if (threadIdx.x < st) red[threadIdx.x] += red[threadIdx.x + st];
        __syncthreads();
    }
    if (threadIdx.x == 0) {
        float inv = 1.0f / (sqrtf(red[0]) + 1e-8f);
        if (row < Sq) rinv[row] = inv; else tinv[row - Sq] = inv;
    }
}

// ---------------------------------------------------------------------------
// NN map: block per tgt token; argmax over masked refs (first-max tiebreak).
// grid 1024, block 256
// ---------------------------------------------------------------------------
__global__ __launch_bounds__(256) void nn_kernel(
    const float* __restrict__ ref, const float* __restrict__ tgt,
    const float* __restrict__ rinv, const float* __restrict__ tinv,
    const unsigned char* __restrict__ mask,
    int* __restrict__ nn_idx, int* __restrict__ keep)
{
    __shared__ float bv[256];
    __shared__ int   bi[256];
    const int t = blockIdx.x;
    const float* trow = tgt + (size_t)t * CDq;
    const float tin = tinv[t];

    float best = -3e38f;
    int   bidx = 0;
    for (int r = (int)threadIdx.x; r < Sq; r += 256) {
        const float* rrow = ref + (size_t)r * CDq;
        float dot = 0.f;
        for (int c = 0; c < CDq; ++c) dot += trow[c] * rrow[c];
        float sim = mask[r] ? (dot * tin * rinv[r]) : -1e9f;
        if (sim > best) { best = sim; bidx = r; }
    }
    bv[threadIdx.x] = best;
    bi[threadIdx.x] = bidx;
    __syncthreads();
    for (int st = 128; st > 0; st >>= 1) {
        if (threadIdx.x < st) {
            float v2 = bv[threadIdx.x + st];
            int   i2 = bi[threadIdx.x + st];
            if (v2 > bv[threadIdx.x] ||
                (v2 == bv[threadIdx.x] && i2 < bi[threadIdx.x])) {
                bv[threadIdx.x] = v2; bi[threadIdx.x] = i2;
            }
        }
        __syncthreads();
    }
    if (threadIdx.x == 0) {
        nn_idx[t] = bi[0];
        keep[t]   = ((1.0f - bv[0]) < 0.7f) ? 1 : 0;
    }
}

// ---------------------------------------------------------------------------
// K/V replacement for batch GEN_COND from batch REF_COND. grid 2560, block 256
// ---------------------------------------------------------------------------
__global__ __launch_bounds__(256) void kv_replace_kernel(
    float* __restrict__ kb, float* __restrict__ vb,
    const int* __restrict__ nn_idx, const int* __restrict__ keep)
{
    int i = blockIdx.x * 256 + threadIdx.x;
    if (i >= Sq * Cq) return;
    int s = i / Cq, c = i - s * Cq;
    if (keep[s]) {
        const size_t gen = (size_t)GEN_COND * Sq * Cq;
        const size_t ref = (size_t)REF_COND * Sq * Cq;
        int ns = nn_idx[s];
        kb[gen + (size_t)s * Cq + c] = kb[ref + (size_t)ns * Cq + c];
        vb[gen + (size_t)s * Cq + c] = vb[ref + (size_t)ns * Cq + c];
    }
}

// ---------------------------------------------------------------------------
// Flash attention: one wave per (b, h, 16-query tile). Scores via WMMA(K=80),
// online softmax with width-16 shuffle reductions, P staged through LDS to
// convert C-layout -> A-layout, P@V via WMMA (K=16), 5 d-chunks of 16.
// grid: (16, 8, 4)  block: 128 (4 waves)
// ---------------------------------------------------------------------------
__global__ __launch_bounds__(128) void attn_kernel(
    const float* __restrict__ qb, const float* __restrict__ kb,
    const float* __restrict__ vb, float* __restrict__ ob)
{
    __shared__ float ldsP[4][16][17];

    const int wave = threadIdx.x >> 5;
    const int lane = threadIdx.x & 31;
    const int lh   = lane & 15;
    const int hf   = lane >> 4;

    const int b  = blockIdx.z;
    const int h  = blockIdx.y;
    const int q0 = (blockIdx.x * 4 + wave) * 16;

    const size_t boff = (size_t)b * Sq * Cq;
    const int    hoff = h * Dq;
    const float  scale = 0.11180339887498949f;   // 1/sqrt(80)

    // Q tile in A-layout: 20 k-steps of 4
    v2f qa[20];
    const float* qp = qb + boff + (size_t)(q0 + lh) * Cq + hoff + 2 * hf;
#pragma unroll
    for (int j = 0; j < 20; ++j) qa[j] = *(const v2f*)(qp + 4 * j);

    v8f o[5];
#pragma unroll
    for (int t = 0; t < 5; ++t) o[t] = (v8f){};
    float mrow[8], lrow[8];
#pragma unroll
    for (int r = 0; r < 8; ++r) { mrow[r] = -3e38f; lrow[r] = 0.f; }

    const float* kbase = kb + boff + hoff;
    const float* vbase = vb + boff + hoff;

    for (int kt = 0; kt < Sq / 16; ++kt) {
        const int key0 = kt * 16;

        // S = Q K^T for this 16x16 tile
        v8f s = {};
        const float* kp = kbase + (size_t)(key0 + lh) * Cq + 2 * hf;
#pragma unroll
        for (int j = 0; j < 20; ++j) {
            v2f kvec = *(const v2f*)(kp + 4 * j);
            s = __builtin_amdgcn_wmma_f32_16x16x4_f32(
                    false, qa[j], false, kvec, (short)0, s, false, false);
        }

        // online softmax; row = r + 8*hf, column = lh within the 16-lane half
        float alpha[8];
#pragma unroll
        for (int r = 0; r < 8; ++r) {
            float v = s[r] * scale;
            float rmax = v;
            rmax = fmaxf(rmax, __shfl_xor(rmax, 1, 32));
            rmax = fmaxf(rmax, __shfl_xor(rmax, 2, 32));
            rmax = fmaxf(rmax, __shfl_xor(rmax, 4, 32));
            rmax = fmaxf(rmax, __shfl_xor(rmax, 8, 32));
            float mnew = fmaxf(mrow[r], rmax);
            float p = __expf(v - mnew);
            float rsum = p;
            rsum += __shfl_xor(rsum, 1, 32);
            rsum += __shfl_xor(rsum, 2, 32);
            rsum += __shfl_xor(rsum, 4, 32);
            rsum += __shfl_xor(rsum, 8, 32);
            alpha[r] = __expf(mrow[r] - mnew);
            lrow[r]  = lrow[r] * alpha[r] + rsum;
            mrow[r]  = mnew;
            ldsP[wave][r + 8 * hf][lh] = p;    // C-layout -> LDS tile stored by row, then column
        }

        // rescale accumulator
#pragma unroll
        for (int t = 0; t < 5; ++t)
#pragma unroll
            for (int r = 0; r < 8; ++r) o[t][r] *= alpha[r];

        // O += P @ V   (in-wave LDS ops are in order; no barrier needed)
#pragma unroll
        for (int j = 0; j < 4; ++j) {
            const int kk = 4 * j + 2 * hf;
            v2f pa;
            pa.x = ldsP[wave][lh][kk];
            pa.y = ldsP[wave][lh][kk + 1];
            const float* vr0 = vbase + (size_t)(key0 + kk) * Cq + lh;
            const float* vr1 = vr0 + Cq;
#pragma unroll
            for (int t = 0; t < 5; ++t) {
                v2f vv;
                vv.x = vr0[t * 16];
                vv.y = vr1[t * 16];
                o[t] = __builtin_amdgcn_wmma_f32_16x16x4_f32(
                           false, pa, false, vv, (short)0, o[t], false, false);
            }
        }
    }

    // normalize and write [b, q, h*80 + d]
    float* op = ob + boff + (size_t)q0 * Cq + hoff;
#pragma unroll
    for (int r = 0; r < 8; ++r) {
        const int row = r + 8 * hf;
        const float inv = 1.0f / lrow[r];
#pragma unroll
        for (int t = 0; t < 5; ++t)
            op[(size_t)row * Cq + t * 16 + lh] = o[t][r] * inv;
    }
}

// ---------------------------------------------------------------------------
// Output projection: out = A @ Wo^T + bo + residual, 32x32 register blocking.
// grid (128, 5)  block 128
// ---------------------------------------------------------------------------
__global__ __launch_bounds__(128) void out_proj_kernel(
    const float* __restrict__ A, const float* __restrict__ Wo,
    const float* __restrict__ bo, const float* __restrict__ resid,
    float* __restrict__ Y)
{
    const int wave = threadIdx.x >> 5;
    const int lane = threadIdx.x & 31;
    const int lh   = lane & 15;
    const int hf   = lane >> 4;

    const int m0 = blockIdx.x * 32;
    const int n0 = (blockIdx.y * 4 + wave) * 32;

    v8f acc00 = {}, acc01 = {}, acc10 = {}, acc11 = {};

    const float* xp0 = A  + (size_t)(m0 + lh) * Cq + 2 * hf;
    const float* xp1 = xp0 + (size_t)16 * Cq;
    const float* wp0 = Wo + (size_t)(n0 + lh) * Cq + 2 * hf;
    const float* wp1 = wp0 + (size_t)16 * Cq;

#pragma unroll 4
    for (int k0 = 0; k0 < Cq; k0 += 4) {
        v2f a0 = *(const v2f*)(xp0 + k0);
        v2f a1 = *(const v2f*)(xp1 + k0);
        v2f b0 = *(const v2f*)(wp0 + k0);
        v2f b1 = *(const v2f*)(wp1 + k0);
        acc00 = __builtin_amdgcn_wmma_f32_16x16x4_f32(
                    false, a0, false, b0, (short)0, acc00, false, false);
        acc01 = __builtin_amdgcn_wmma_f32_16x16x4_f32(
                    false, a0, false, b1, (short)0, acc01, false, false);
        acc10 = __builtin_amdgcn_wmma_f32_16x16x4_f32(
                    false, a1, false, b0, (short)0, acc10, false, false);
        acc11 = __builtin_amdgcn_wmma_f32_16x16x4_f32(
                    false, a1, false, b1, (short)0, acc11, false, false);
    }

    const float bias0 = bo[n0 + lh];
    const float bias1 = bo[n0 + 16 + lh];
#pragma unroll
    for (int r = 0; r < 8; ++r) {
        int row = r + 8 * hf;
        size_t i00 = (size_t)(m0 + row)      * Cq + n0 + lh;
        size_t i01 = (size_t)(m0 + row)      * Cq + n0 + 16 + lh;
        size_t i10 = (size_t)(m0 + 16 + row) * Cq + n0 + lh;
        size_t i11 = (size_t)(m0 + 16 + row) * Cq + n0 + 16 + lh;
        Y[i00] = acc00[r] + bias0 + resid[i00];
        Y[i01] = acc01[r] + bias1 + resid[i01];
        Y[i10] = acc10[r] + bias0 + resid[i10];
        Y[i11] = acc11[r] + bias1 + resid[i11];
    }
}

// ---------------------------------------------------------------------------
extern "C" void kernel_launch(void* const* d_in, const int* in_sizes, int n_in,
                              void* d_out, int out_size, void* d_ws, size_t ws_size,
                              hipStream_t stream)
{
    (void)in_sizes; (void)n_in; (void)out_size; (void)ws_size;

    const float* hidden = (const float*)d_in[0];
    const float* Wq     = (const float*)d_in[1];
    const float* Wk     = (const float*)d_in[2];
    const float* Wv     = (const float*)d_in[3];
    const float* Wo     = (const float*)d_in[4];
    const float* bo     = (const float*)d_in[5];
    const float* refd   = (const float*)d_in[6];
    const float* tgtd   = (const float*)d_in[7];
    const unsigned char* ref_mask = (const unsigned char*)d_in[8];
    // d_in[9] = heads (==8), compile-time constant here

    const size_t sz = (size_t)Bq * Sq * Cq;     // 2,621,440 floats
    float* ws   = (float*)d_ws;
    float* qbuf = ws;
    float* kbuf = ws + sz;
    float* vbuf = ws + 2 * sz;
    float* abuf = ws + 3 * sz;
    float* rinv = ws + 4 * sz;
    float* tinv = rinv + Sq;
    int*   nn_i = (int*)(tinv + Sq);
    int*   keep = nn_i + Sq;

    // 32x32 blocks: M=4096 -> 128, N=640 -> 20 n-blocks = grid.y(5) * 4 waves
    qkv_proj_kernel<<<dim3(Sq * Bq / 32, 5, 3), 128, 0, stream>>>(
        hidden, Wq, Wk, Wv, qbuf, kbuf, vbuf);

    inv_norm_kernel<<<2 * Sq, 256, 0, stream>>>(refd, tgtd, rinv, tinv);

    nn_kernel<<<Sq, 256, 0, stream>>>(refd, tgtd, rinv, tinv, ref_mask, nn_i, keep);

    kv_replace_kernel<<<(Sq * Cq + 255) / 256, 256, 0, stream>>>(kbuf, vbuf, nn_i, keep);

    attn_kernel<<<dim3(Sq / 64, Hq, Bq), 128, 0, stream>>>(qbuf, kbuf, vbuf, abuf);

    out_proj_kernel<<<dim3(Sq * Bq / 32, 5), 128, 0, stream>>>(
        abuf, Wo, bo, hidden, (float*)d_out);
}